// HierarchicalGNN_73787538145947
// MI455X (gfx1250) — compile-verified
//
#include <hip/hip_runtime.h>
#include <hip/hip_bf16.h>
#include <math.h>

#define NMIC 400000
#define EMIC 3200000
#define NMAC 20000
#define EMAC 160000

typedef float v2f __attribute__((ext_vector_type(2)));
typedef float v8f __attribute__((ext_vector_type(8)));

static __device__ __forceinline__ void atomicMaxF(float* addr, float val) {
    int* iaddr = (int*)addr;
    int old = __float_as_int(*addr);
    while (__int_as_float(old) < val) {
        int assumed = old;
        old = atomicCAS(iaddr, assumed, __float_as_int(val));
        if (old == assumed) break;
    }
}

// ---------------------------------------------------------------------------
// fp32 WMMA GEMM: C[M,N] = A[M,K] @ B[K,N] (+bias) (+relu)
// Compile-time K, N, NT. One wave computes a 16 x (16*NT) strip of C.
// Requires M % 16 == 0. With N/NT compile-time, all B masks fold away except
// the genuinely partial N=8 case; K-tail masks remain only in the last step
// of K%4!=0 shapes. Loads are unconditional (clamped addresses).
// ---------------------------------------------------------------------------
template <int K, int N, int NT>
__global__ void hgnn_gemm_wmma(const float* __restrict__ A, int lda,
                               const float* __restrict__ B, int ldb,
                               const float* __restrict__ bias,
                               float* __restrict__ C, int ldc,
                               int M, int act) {
    constexpr int NSUP = (N + 16 * NT - 1) / (16 * NT);
    int tile = blockIdx.x * (blockDim.x >> 5) + (threadIdx.x >> 5);
    int total = (M >> 4) * NSUP;
    if (tile >= total) return;                 // wave-uniform exit
    int m0 = (tile / NSUP) << 4;
    int n0 = (tile % NSUP) * (16 * NT);
    int lane = threadIdx.x & 31;
    int half = lane >> 4;                      // lanes 0-15 vs 16-31
    int l16  = lane & 15;

    const float* Arow = A + (size_t)(m0 + l16) * lda;

    v8f acc[NT] = {};
    float bmask[NT];
    int   bcol[NT];
#pragma unroll
    for (int t = 0; t < NT; ++t) {
        int col  = n0 + t * 16 + l16;
        bmask[t] = (col < N) ? 1.0f : 0.0f;    // folds to 1.0 when N % (16*NT) == 0
        bcol[t]  = (col < N) ? col : (N - 1);
    }

    constexpr int KSTEPS = (K + 3) / 4;
#pragma unroll
    for (int ks = 0; ks < KSTEPS; ++ks) {
        int ka  = ks * 4 + half * 2;           // lanes 0-15: K=4ks,4ks+1 ; 16-31: +2,+3
        float km0 = (ka     < K) ? 1.0f : 0.0f;
        float km1 = (ka + 1 < K) ? 1.0f : 0.0f;
        int k0c = (ka     < K) ? ka     : (K - 1);
        int k1c = (ka + 1 < K) ? ka + 1 : (K - 1);
        v2f a;
        a.x = Arow[k0c] * km0;
        a.y = Arow[k1c] * km1;
        const float* B0 = B + (size_t)k0c * ldb;
        const float* B1 = B + (size_t)k1c * ldb;
#pragma unroll
        for (int t = 0; t < NT; ++t) {
            v2f b;
            b.x = B0[bcol[t]] * (km0 * bmask[t]);
            b.y = B1[bcol[t]] * (km1 * bmask[t]);
            acc[t] = __builtin_amdgcn_wmma_f32_16x16x4_f32(false, a, false, b,
                                                           (short)0, acc[t], false, false);
        }
    }

#pragma unroll
    for (int t = 0; t < NT; ++t) {
        int col = n0 + t * 16 + l16;
        if (col < N) {
            float bv = bias ? bias[col] : 0.0f;
#pragma unroll
            for (int i = 0; i < 8; ++i) {
                int row = m0 + (half << 3) + i;   // VGPR i: rows i / i+8
                float v = acc[t][i] + bv;
                if (act == 1) v = fmaxf(v, 0.0f);
                C[(size_t)row * ldc + col] = v;
            }
        }
    }
}

// ---------------------------------------------------------------------------
// Elementwise / graph kernels
// ---------------------------------------------------------------------------
__global__ void hgnn_fill(float* p, long long n, float v) {
    long long i = (long long)blockIdx.x * blockDim.x + threadIdx.x;
    if (i < n) p[i] = v;
}

__global__ void hgnn_deg_count(const int* __restrict__ dst, int E, float* deg) {
    int e = blockIdx.x * blockDim.x + threadIdx.x;
    if (e < E) atomicAdd(&deg[dst[e]], 1.0f);
}

__global__ void hgnn_rsqrt_ip(float* p, int n) {
    int i = blockIdx.x * blockDim.x + threadIdx.x;
    if (i < n) p[i] = rsqrtf(p[i]);
}

// agg[dst, c..c+3] += h[src, c..c+3] * dinv[src]*dinv[dst]   (float4 per thread)
__global__ void hgnn_gcn_agg(const float* __restrict__ h, const int* __restrict__ src,
                             const int* __restrict__ dst, const float* __restrict__ dinv,
                             float* agg, int E) {
    long long idx = (long long)blockIdx.x * blockDim.x + threadIdx.x;
    if (idx >= (long long)E * 16) return;
    int e  = (int)(idx >> 4);
    int c4 = ((int)idx & 15) * 4;
    int s = src[e], d = dst[e];
    float nrm = dinv[s] * dinv[d];
    float4 hv = *(const float4*)(h + (size_t)s * 64 + c4);
    float* o = agg + (size_t)d * 64 + c4;
    atomicAdd(o + 0, hv.x * nrm);
    atomicAdd(o + 1, hv.y * nrm);
    atomicAdd(o + 2, hv.z * nrm);
    atomicAdd(o + 3, hv.w * nrm);
}

// x = relu(bn2(relu(bn1(agg + h*dinv^2 + b1))))
__global__ void hgnn_gcn1_post(const float* __restrict__ agg, const float* __restrict__ h,
                               const float* __restrict__ dinv,
                               const float* b1, const float* g1, const float* be1,
                               const float* g2, const float* be2, float* out, long long n) {
    long long idx = (long long)blockIdx.x * blockDim.x + threadIdx.x;
    if (idx >= n) return;
    int node = (int)(idx >> 6), c = (int)(idx & 63);
    float di = dinv[node];
    float v = agg[idx] + h[idx] * di * di + b1[c];
    float inv = rsqrtf(1.0f + 1e-5f);
    v = fmaxf(v * (g1[c] * inv) + be1[c], 0.0f);
    v = fmaxf(v * (g2[c] * inv) + be2[c], 0.0f);
    out[idx] = v;
}

__global__ void hgnn_gcn2_post(const float* __restrict__ agg, const float* __restrict__ h,
                               const float* __restrict__ dinv, const float* b2,
                               float* out, long long n) {
    long long idx = (long long)blockIdx.x * blockDim.x + threadIdx.x;
    if (idx >= n) return;
    int node = (int)(idx >> 6), c = (int)(idx & 63);
    float di = dinv[node];
    out[idx] = agg[idx] + h[idx] * di * di + b2[c];
}

__global__ void hgnn_pool_cnt(const int* __restrict__ batch, int n, float* cnt) {
    int i = blockIdx.x * blockDim.x + threadIdx.x;
    if (i < n) atomicAdd(&cnt[batch[i]], 1.0f);
}

// float4 per thread: 16 threads per node
__global__ void hgnn_pool(const float* __restrict__ x, const int* __restrict__ batch,
                          float* psum, float* pmax, long long n16) {
    long long idx = (long long)blockIdx.x * blockDim.x + threadIdx.x;
    if (idx >= n16) return;
    int node = (int)(idx >> 4);
    int c4 = ((int)idx & 15) * 4;
    int b = batch[node];
    float4 v = *(const float4*)(x + (size_t)node * 64 + c4);
    float* ps = psum + (size_t)b * 64 + c4;
    float* pm = pmax + (size_t)b * 64 + c4;
    atomicAdd(ps + 0, v.x); atomicAdd(ps + 1, v.y);
    atomicAdd(ps + 2, v.z); atomicAdd(ps + 3, v.w);
    atomicMaxF(pm + 0, v.x); atomicMaxF(pm + 1, v.y);
    atomicMaxF(pm + 2, v.z); atomicMaxF(pm + 3, v.w);
}

__global__ void hgnn_zfin(const float* __restrict__ psum, const float* __restrict__ pmax,
                          const float* __restrict__ cnt, float* z, int n) {
    int idx = blockIdx.x * blockDim.x + threadIdx.x;
    if (idx >= n) return;
    int m = idx >> 6, c = idx & 63;
    float cn = cnt[m];
    float mean = psum[idx] / fmaxf(cn, 1.0f);
    float mx = (cn > 0.0f) ? pmax[idx] : 0.0f;
    z[(size_t)m * 128 + c] = mean;
    z[(size_t)m * 128 + 64 + c] = mx;
}

__global__ void hgnn_concat_xc(const float* __restrict__ zm, const float* __restrict__ xmac,
                               float* xc, int n) {
    int idx = blockIdx.x * blockDim.x + threadIdx.x;
    if (idx >= n) return;
    int m = idx / 34, j = idx % 34;
    xc[idx] = (j < 32) ? zm[(size_t)m * 32 + j] : xmac[(size_t)m * 2 + (j - 32)];
}

__global__ void hgnn_eself_acc(const int* __restrict__ dst, const float* __restrict__ ea,
                               float* eself, int E) {
    int idx = blockIdx.x * blockDim.x + threadIdx.x;
    if (idx >= E * 4) return;
    int e = idx >> 2, k = idx & 3;
    atomicAdd(&eself[(size_t)dst[e] * 4 + k], ea[idx]);
}

__global__ void hgnn_eself_div(float* eself, const float* __restrict__ cnt, int n) {
    int idx = blockIdx.x * blockDim.x + threadIdx.x;
    if (idx >= n) return;
    eself[idx] /= fmaxf(cnt[idx >> 2], 1.0f);
}

// GATv2 pass 1: score[e2,h] = sum_c leaky(xl[s]+xr[d]+ea2@We) * att ; smax = segmax over dst
__global__ void hgnn_gat_score(const float* __restrict__ xl, const float* __restrict__ xr,
                               const float* __restrict__ We, const float* __restrict__ att,
                               const int* __restrict__ src, const int* __restrict__ dst,
                               const float* __restrict__ ea, const float* __restrict__ eself,
                               float* score, float* smax, int E, int Nn, int heads, int oc) {
    int idx = blockIdx.x * blockDim.x + threadIdx.x;
    int E2 = E + Nn;
    if (idx >= E2 * heads) return;
    int e = idx / heads, h = idx % heads;
    int s, d;
    const float* eav;
    if (e < E) { s = src[e]; d = dst[e]; eav = ea + (size_t)e * 4; }
    else       { s = d = e - E;          eav = eself + (size_t)(e - E) * 4; }
    int HO = heads * oc;
    float e0 = eav[0], e1 = eav[1], e2v = eav[2], e3 = eav[3];
    const float* xls = xl + (size_t)s * HO + h * oc;
    const float* xrd = xr + (size_t)d * HO + h * oc;
    const float* wc  = We + h * oc;          // We is [4][HO]
    const float* av  = att + h * oc;
    float acc = 0.0f;
    for (int c = 0; c < oc; ++c) {
        float em = e0 * wc[c] + e1 * wc[HO + c] + e2v * wc[2 * HO + c] + e3 * wc[3 * HO + c];
        float m = xls[c] + xrd[c] + em;
        m = (m > 0.0f) ? m : 0.2f * m;
        acc += m * av[c];
    }
    score[idx] = acc;
    atomicMaxF(&smax[d * heads + h], acc);
}

// GATv2 pass 2: ex = exp(score - smax[d]); den[d] += ex
__global__ void hgnn_gat_exp(const int* __restrict__ dst, float* score,
                             const float* __restrict__ smax, float* den,
                             int E, int Nn, int heads) {
    int idx = blockIdx.x * blockDim.x + threadIdx.x;
    int E2 = E + Nn;
    if (idx >= E2 * heads) return;
    int e = idx / heads, h = idx % heads;
    int d = (e < E) ? dst[e] : e - E;
    float ex = expf(score[idx] - smax[d * heads + h]);
    score[idx] = ex;
    atomicAdd(&den[d * heads + h], ex);
}

// GATv2 pass 3: out[d, j..j+3] += xl[s, j..j+3] * (ex / den[d,h])   (float4 per thread)
__global__ void hgnn_gat_agg(const float* __restrict__ xl, const float* __restrict__ score,
                             const float* __restrict__ den, const int* __restrict__ src,
                             const int* __restrict__ dst, float* out,
                             int E, int Nn, int heads, int oc) {
    long long idx = (long long)blockIdx.x * blockDim.x + threadIdx.x;
    int HO = heads * oc;
    int q = HO >> 2;                         // float4 groups per node
    long long total = (long long)(E + Nn) * q;
    if (idx >= total) return;
    int e  = (int)(idx / q);
    int j4 = ((int)(idx % q)) * 4;
    int h  = j4 / oc;
    int s, d;
    if (e < E) { s = src[e]; d = dst[e]; } else { s = d = e - E; }
    float alpha = score[(size_t)e * heads + h] / den[d * heads + h];
    float4 xv = *(const float4*)(xl + (size_t)s * HO + j4);
    float* o = out + (size_t)d * HO + j4;
    atomicAdd(o + 0, xv.x * alpha);
    atomicAdd(o + 1, xv.y * alpha);
    atomicAdd(o + 2, xv.z * alpha);
    atomicAdd(o + 3, xv.w * alpha);
}

// act: 0 none, 2 elu
__global__ void hgnn_bias_act(float* x, const float* __restrict__ b, long long n,
                              int cols, int act) {
    long long idx = (long long)blockIdx.x * blockDim.x + threadIdx.x;
    if (idx >= n) return;
    float v = x[idx] + b[(int)(idx % cols)];
    if (act == 2) v = (v > 0.0f) ? v : (expf(v) - 1.0f);
    x[idx] = v;
}

// ---------------------------------------------------------------------------
// Host-side orchestration
// ---------------------------------------------------------------------------
static inline int cdiv(long long a, int b) { return (int)((a + b - 1) / b); }

#define GEMM_LAUNCH(KC, NC, NT, A, lda, B, ldb, bias, C, ldc, M, act)                  \
    do {                                                                               \
        int _tiles = ((M) / 16) * (((NC) + 16 * (NT) - 1) / (16 * (NT)));              \
        hgnn_gemm_wmma<KC, NC, NT><<<(_tiles + 7) / 8, 256, 0, stream>>>(              \
            A, lda, B, ldb, bias, C, ldc, M, act);                                     \
    } while (0)

extern "C" void kernel_launch(void* const* d_in, const int* in_sizes, int n_in,
                              void* d_out, int out_size, void* d_ws, size_t ws_size,
                              hipStream_t stream) {
    const float* x_micro = (const float*)d_in[0];
    const int*   ei_mic  = (const int*)d_in[1];
    const int*   batch   = (const int*)d_in[2];
    const float* x_macro = (const float*)d_in[3];
    const int*   ei_mac  = (const int*)d_in[4];
    const float* ea_mac  = (const float*)d_in[5];
    const float *W1 = (const float*)d_in[6],  *b1 = (const float*)d_in[7];
    const float *g1 = (const float*)d_in[8],  *be1 = (const float*)d_in[9];
    const float *g2 = (const float*)d_in[10], *be2 = (const float*)d_in[11];
    const float *W2 = (const float*)d_in[12], *b2 = (const float*)d_in[13];
    const float *roW1 = (const float*)d_in[14], *rob1 = (const float*)d_in[15];
    const float *roW2 = (const float*)d_in[16], *rob2 = (const float*)d_in[17];
    const float *g1Wl = (const float*)d_in[18], *g1Wr = (const float*)d_in[19];
    const float *g1We = (const float*)d_in[20], *g1att = (const float*)d_in[21];
    const float *g1b  = (const float*)d_in[22];
    const float *g2Wl = (const float*)d_in[23], *g2Wr = (const float*)d_in[24];
    const float *g2We = (const float*)d_in[25], *g2att = (const float*)d_in[26];
    const float *g2b  = (const float*)d_in[27];
    const float *dW1 = (const float*)d_in[28], *db1 = (const float*)d_in[29];
    const float *dW2 = (const float*)d_in[30], *db2 = (const float*)d_in[31];

    const int* src_mic = ei_mic;
    const int* dst_mic = ei_mic + EMIC;
    const int* src_mac = ei_mac;
    const int* dst_mac = ei_mac + EMAC;

    float* ws   = (float*)d_ws;
    float* bufA = ws;                         // 25,600,000 f32
    float* bufB = ws + 25600000;              // 25,600,000 f32
    float* deg  = ws + 51200000;              // 400,000 f32 (becomes dinv)

    // macro buffers overlaid into bufB (free after GCN2 finalize)
    float* psum = bufB;                       // 1,280,000
    float* pmax = bufB + 1280000;             // 1,280,000
    float* pcnt = bufB + 2560000;             // 20,000
    float* z    = bufB + 2580000;             // 2,560,000
    // macro buffers overlaid into bufA (free after pooling)
    float* rohid = bufA;                      // 1,280,000
    float* zm    = bufA + 1280000;            // 640,000
    float* xc    = bufA + 1920000;            // 680,000
    float* eself = bufA + 2600000;            // 80,000
    float* cntm  = bufA + 2680000;            // 20,000
    float* xl1   = bufA + 2700000;            // 5,120,000
    float* xr1   = bufA + 7820000;            // 5,120,000
    float* g1out = bufA + 12940000;           // 5,120,000
    float* score = bufA + 18060000;           // 720,000
    float* smax  = bufA + 18780000;           // 80,000
    float* den   = bufA + 18860000;           // 80,000
    float* xl2   = bufA + 18940000;           // 640,000
    float* xr2   = bufA + 19580000;           // 640,000
    float* zfin  = bufA + 20220000;           // 640,000
    float* dhid  = bufA + 20860000;           // 1,280,000 (ends 22,140,000)

    const long long NM64 = (long long)NMIC * 64;

    // ---------------- micro GCN layer 1 ----------------
    GEMM_LAUNCH(17, 64, 4, x_micro, 17, W1, 64, nullptr, bufA, 64, NMIC, 0);   // h1
    hgnn_fill<<<cdiv(NMIC, 256), 256, 0, stream>>>(deg, NMIC, 1.0f);
    hgnn_deg_count<<<cdiv(EMIC, 256), 256, 0, stream>>>(dst_mic, EMIC, deg);
    hgnn_rsqrt_ip<<<cdiv(NMIC, 256), 256, 0, stream>>>(deg, NMIC);             // deg -> dinv
    hgnn_fill<<<cdiv(NM64, 256), 256, 0, stream>>>(bufB, NM64, 0.0f);
    hgnn_gcn_agg<<<cdiv((long long)EMIC * 16, 256), 256, 0, stream>>>(bufA, src_mic, dst_mic,
                                                                      deg, bufB, EMIC);
    hgnn_gcn1_post<<<cdiv(NM64, 256), 256, 0, stream>>>(bufB, bufA, deg, b1, g1, be1,
                                                        g2, be2, bufA, NM64);
    // ---------------- micro GCN layer 2 ----------------
    GEMM_LAUNCH(64, 64, 4, bufA, 64, W2, 64, nullptr, bufB, 64, NMIC, 0);      // h2
    hgnn_fill<<<cdiv(NM64, 256), 256, 0, stream>>>(bufA, NM64, 0.0f);
    hgnn_gcn_agg<<<cdiv((long long)EMIC * 16, 256), 256, 0, stream>>>(bufB, src_mic, dst_mic,
                                                                      deg, bufA, EMIC);
    hgnn_gcn2_post<<<cdiv(NM64, 256), 256, 0, stream>>>(bufA, bufB, deg, b2, bufA, NM64);
    // ---------------- pooling ----------------
    hgnn_fill<<<cdiv(NMAC, 256), 256, 0, stream>>>(pcnt, NMAC, 0.0f);
    hgnn_fill<<<cdiv(NMAC * 64, 256), 256, 0, stream>>>(psum, NMAC * 64, 0.0f);
    hgnn_fill<<<cdiv(NMAC * 64, 256), 256, 0, stream>>>(pmax, NMAC * 64, -3.402823466e38f);
    hgnn_pool_cnt<<<cdiv(NMIC, 256), 256, 0, stream>>>(batch, NMIC, pcnt);
    hgnn_pool<<<cdiv((long long)NMIC * 16, 256), 256, 0, stream>>>(bufA, batch, psum, pmax,
                                                                   (long long)NMIC * 16);
    hgnn_zfin<<<cdiv(NMAC * 64, 256), 256, 0, stream>>>(psum, pmax, pcnt, z, NMAC * 64);
    // ---------------- readout MLP ----------------
    GEMM_LAUNCH(128, 64, 4, z, 128, roW1, 64, rob1, rohid, 64, NMAC, 1);
    GEMM_LAUNCH(64, 32, 2, rohid, 64, roW2, 32, rob2, zm, 32, NMAC, 0);
    hgnn_concat_xc<<<cdiv(NMAC * 34, 256), 256, 0, stream>>>(zm, x_macro, xc, NMAC * 34);
    // ---------------- GATv2 layer 1 (heads=4, oc=64, concat) ----------------
    hgnn_fill<<<cdiv(NMAC, 256), 256, 0, stream>>>(cntm, NMAC, 0.0f);
    hgnn_fill<<<cdiv(NMAC * 4, 256), 256, 0, stream>>>(eself, NMAC * 4, 0.0f);
    hgnn_deg_count<<<cdiv(EMAC, 256), 256, 0, stream>>>(dst_mac, EMAC, cntm);
    hgnn_eself_acc<<<cdiv(EMAC * 4, 256), 256, 0, stream>>>(dst_mac, ea_mac, eself, EMAC);
    hgnn_eself_div<<<cdiv(NMAC * 4, 256), 256, 0, stream>>>(eself, cntm, NMAC * 4);
    GEMM_LAUNCH(34, 256, 4, xc, 34, g1Wl, 256, nullptr, xl1, 256, NMAC, 0);
    GEMM_LAUNCH(34, 256, 4, xc, 34, g1Wr, 256, nullptr, xr1, 256, NMAC, 0);
    hgnn_fill<<<cdiv(NMAC * 4, 256), 256, 0, stream>>>(smax, NMAC * 4, -3.402823466e38f);
    hgnn_fill<<<cdiv(NMAC * 4, 256), 256, 0, stream>>>(den, NMAC * 4, 0.0f);
    hgnn_fill<<<cdiv(NMAC * 256, 256), 256, 0, stream>>>(g1out, NMAC * 256, 0.0f);
    {
        int E2h = (EMAC + NMAC) * 4;
        hgnn_gat_score<<<cdiv(E2h, 128), 128, 0, stream>>>(xl1, xr1, g1We, g1att, src_mac,
                                                           dst_mac, ea_mac, eself, score, smax,
                                                           EMAC, NMAC, 4, 64);
        hgnn_gat_exp<<<cdiv(E2h, 256), 256, 0, stream>>>(dst_mac, score, smax, den,
                                                         EMAC, NMAC, 4);
        hgnn_gat_agg<<<cdiv((long long)(EMAC + NMAC) * 64, 256), 256, 0, stream>>>(
            xl1, score, den, src_mac, dst_mac, g1out, EMAC, NMAC, 4, 64);
        hgnn_bias_act<<<cdiv((long long)NMAC * 256, 256), 256, 0, stream>>>(g1out, g1b,
                                                                 (long long)NMAC * 256, 256, 2);
    }
    // ---------------- GATv2 layer 2 (heads=1, oc=32, mean==identity) ----------------
    GEMM_LAUNCH(256, 32, 2, g1out, 256, g2Wl, 32, nullptr, xl2, 32, NMAC, 0);
    GEMM_LAUNCH(256, 32, 2, g1out, 256, g2Wr, 32, nullptr, xr2, 32, NMAC, 0);
    hgnn_fill<<<cdiv(NMAC, 256), 256, 0, stream>>>(smax, NMAC, -3.402823466e38f);
    hgnn_fill<<<cdiv(NMAC, 256), 256, 0, stream>>>(den, NMAC, 0.0f);
    hgnn_fill<<<cdiv(NMAC * 32, 256), 256, 0, stream>>>(zfin, NMAC * 32, 0.0f);
    {
        int E2h = (EMAC + NMAC) * 1;
        hgnn_gat_score<<<cdiv(E2h, 128), 128, 0, stream>>>(xl2, xr2, g2We, g2att, src_mac,
                                                           dst_mac, ea_mac, eself, score, smax,
                                                           EMAC, NMAC, 1, 32);
        hgnn_gat_exp<<<cdiv(E2h, 256), 256, 0, stream>>>(dst_mac, score, smax, den,
                                                         EMAC, NMAC, 1);
        hgnn_gat_agg<<<cdiv((long long)(EMAC + NMAC) * 8, 256), 256, 0, stream>>>(
            xl2, score, den, src_mac, dst_mac, zfin, EMAC, NMAC, 1, 32);
        hgnn_bias_act<<<cdiv((long long)NMAC * 32, 256), 256, 0, stream>>>(zfin, g2b,
                                                                 (long long)NMAC * 32, 32, 0);
    }
    // ---------------- decoder MLP ----------------
    GEMM_LAUNCH(32, 64, 4, zfin, 32, dW1, 64, db1, dhid, 64, NMAC, 1);
    GEMM_LAUNCH(64, 8, 1, dhid, 64, dW2, 8, db2, (float*)d_out, 8, NMAC, 0);
}